// GATModel_20899310862411
// MI455X (gfx1250) — compile-verified
//
#include <hip/hip_runtime.h>
#include <hip/hip_bf16.h>

// ------------------------------------------------------------------
// GAT + SAGPool forward for MI455X (gfx1250, wave32).
// Dense GEMMs -> V_WMMA_F32_16X16X4_F32, N-blocked 4x per wave so the
// A fragment is reused across 4 independent WMMA accumulators.
// All GEMM N dims are multiples of 64 -> unconditional 4-tile strip,
// no EXEC manipulation around WMMA (ISA requires EXEC all-1s).
// Scatter softmax / aggregation -> edge-parallel atomics (L2 resident).
// ------------------------------------------------------------------

#define B_GR   32
#define NPG0   256
#define NE     32768
#define NHEADS 8
#define NEG_BIG (-1e30f)

typedef __attribute__((ext_vector_type(2))) float v2f;
typedef __attribute__((ext_vector_type(8))) float v8f;

enum { EPI_STORE = 0, EPI_LEAKY = 1, EPI_BN_LEAKY = 2, EPI_ACCUM = 3 };

// ---------------- generic WMMA f32 GEMM: C[M,N] = A[M,K] @ W[K,N] (+epilogue)
// one wave per 16(M) x 64(N) strip: 4 accumulators reuse one A fragment.
// 4 waves per block -> block covers 16 x 256. REQUIRES N % 64 == 0.
__global__ void wmma_gemm_f32(const float* __restrict__ A, const float* __restrict__ W,
                              float* __restrict__ C, int M, int N, int K,
                              const float* __restrict__ bias,
                              const float* __restrict__ bn_g, const float* __restrict__ bn_b,
                              const float* __restrict__ bn_m, const float* __restrict__ bn_v,
                              int epi, float slope)
{
    const int wave = threadIdx.x >> 5;
    const int lane = threadIdx.x & 31;
    const int tilesN = N >> 4;                     // 16-col tiles (multiple of 4)
    const int tBase = (blockIdx.x * 4 + wave) * 4; // first 16-col tile of this wave
    const int tM = blockIdx.y;
    if (tBase >= tilesN) return;                   // wave-uniform; EXEC all-1 below

    const int m0 = tM << 4;
    const int half = lane >> 4;                    // 0 or 1
    const int mrow = lane & 15;                    // A row (M)
    const int ncol = lane & 15;                    // B/C col within tile
    const int kh = half << 1;                      // K sub-offset 0 or 2

    v8f acc0 = {}, acc1 = {}, acc2 = {}, acc3 = {};
    const float* __restrict__ Arow = A + (size_t)(m0 + mrow) * K;
    const int nb = (tBase << 4) + ncol;            // col of acc0's lane

    for (int k = 0; k < K; k += 4) {
        v2f a;
        a.x = Arow[k + kh];
        a.y = Arow[k + kh + 1];
        const float* __restrict__ W0 = W + (size_t)(k + kh) * N + nb;
        const float* __restrict__ W1 = W0 + N;
        v2f b0, b1, b2, b3;
        b0.x = W0[0];   b0.y = W1[0];
        b1.x = W0[16];  b1.y = W1[16];
        b2.x = W0[32];  b2.y = W1[32];
        b3.x = W0[48];  b3.y = W1[48];
        acc0 = __builtin_amdgcn_wmma_f32_16x16x4_f32(false, a, false, b0, (short)0, acc0, false, false);
        acc1 = __builtin_amdgcn_wmma_f32_16x16x4_f32(false, a, false, b1, (short)0, acc1, false, false);
        acc2 = __builtin_amdgcn_wmma_f32_16x16x4_f32(false, a, false, b2, (short)0, acc2, false, false);
        acc3 = __builtin_amdgcn_wmma_f32_16x16x4_f32(false, a, false, b3, (short)0, acc3, false, false);
    }

    const int rbase = half << 3;                   // lanes 16-31 hold rows M=8..15
    v8f accs[4] = {acc0, acc1, acc2, acc3};
    for (int j = 0; j < 4; ++j) {
        const int c = nb + j * 16;
        for (int v = 0; v < 8; ++v) {
            int r = m0 + rbase + v;
            float xv = accs[j][v];
            if (bias) xv += bias[c];
            if (epi == EPI_BN_LEAKY) {
                xv = (xv - bn_m[c]) * rsqrtf(bn_v[c] + 1e-5f) * bn_g[c] + bn_b[c];
                xv = xv >= 0.f ? xv : 0.01f * xv;
            } else if (epi == EPI_LEAKY) {
                xv = xv >= 0.f ? xv : slope * xv;
            } else if (epi == EPI_ACCUM) {
                xv += C[(size_t)r * N + c];
            }
            C[(size_t)r * N + c] = xv;
        }
    }
}

// ---------------- fills
__global__ void k_fill_f32(float* p, float v, int n) {
    int t = blockIdx.x * blockDim.x + threadIdx.x; if (t < n) p[t] = v;
}
__global__ void k_fill_i32(int* p, int v, int n) {
    int t = blockIdx.x * blockDim.x + threadIdx.x; if (t < n) p[t] = v;
}

// monotone order-preserving float->uint key (for atomic float max via uint max)
__device__ __forceinline__ unsigned f2key(float f) {
    unsigned u = __float_as_uint(f);
    return (u & 0x80000000u) ? ~u : (u | 0x80000000u);
}
__device__ __forceinline__ float key2f(unsigned k) {
    return (k & 0x80000000u) ? __uint_as_float(k & 0x7fffffffu) : __uint_as_float(~k);
}
__global__ void k_fill_key(unsigned* p, float v, int n) {
    int t = blockIdx.x * blockDim.x + threadIdx.x; if (t < n) p[t] = f2key(v);
}

__global__ void k_copy_edges(const int* __restrict__ eidx, int* src, int* dst, int* emask) {
    int t = blockIdx.x * blockDim.x + threadIdx.x;
    if (t < NE) { src[t] = eidx[t]; dst[t] = eidx[NE + t]; emask[t] = 1; }
}

// ---------------- attention coefficients: asrc/adst[n,h] = <hh[n,h,:], att[h,:]>
__global__ void k_att_coeff(const float* __restrict__ hh, const float* __restrict__ att_s,
                            const float* __restrict__ att_d, float* __restrict__ asrc,
                            float* __restrict__ adst, int Nn, int co)
{
    int t = blockIdx.x * blockDim.x + threadIdx.x;
    if (t >= Nn * NHEADS) return;
    int n = t / NHEADS, h = t % NHEADS;
    const float* hv = hh + ((size_t)n * NHEADS + h) * co;
    const float* as = att_s + (size_t)h * co;
    const float* ad = att_d + (size_t)h * co;
    float s = 0.f, d = 0.f;
    for (int c = 0; c < co; ++c) { float v = hv[c]; s += v * as[c]; d += v * ad[c]; }
    asrc[t] = s; adst[t] = d;
}

// edge logits (with appended self-loops) + segment max via uint-key atomics
__global__ void k_edge_scores(const int* __restrict__ src, const int* __restrict__ dst,
                              const int* __restrict__ emask,
                              const float* __restrict__ asrc, const float* __restrict__ adst,
                              float* __restrict__ ebuf, unsigned* __restrict__ mxk,
                              int Ereal, int Nn)
{
    int t = blockIdx.x * blockDim.x + threadIdx.x;
    int Et = Ereal + Nn;
    if (t >= Et * NHEADS) return;
    int e = t / NHEADS, h = t % NHEADS;
    int s, d, m;
    if (e < Ereal) { s = src[e]; d = dst[e]; m = emask[e]; }
    else           { s = d = e - Ereal; m = 1; }
    float v = NEG_BIG;
    if (m) {
        v = asrc[s * NHEADS + h] + adst[d * NHEADS + h];
        v = v >= 0.f ? v : 0.2f * v;                       // leaky 0.2
        atomicMax(&mxk[d * NHEADS + h], f2key(v));
    }
    ebuf[t] = v;
}

// w = exp(e - mx[d]);  den[d,h] += w   (ebuf overwritten in place)
__global__ void k_edge_w(const int* __restrict__ src, const int* __restrict__ dst,
                         const int* __restrict__ emask, const unsigned* __restrict__ mxk,
                         float* __restrict__ ebuf, float* __restrict__ den, int Ereal, int Nn)
{
    int t = blockIdx.x * blockDim.x + threadIdx.x;
    int Et = Ereal + Nn;
    if (t >= Et * NHEADS) return;
    int e = t / NHEADS, h = t % NHEADS;
    int d, m;
    if (e < Ereal) { d = dst[e]; m = emask[e]; }
    else           { d = e - Ereal; m = 1; }
    float w = 0.f;
    if (m) {
        w = __expf(ebuf[t] - key2f(mxk[d * NHEADS + h]));
        if (w != 0.f) atomicAdd(&den[d * NHEADS + h], w);
    }
    ebuf[t] = w;
}

// out[d,h,:] += alpha * hh[s,h,:]
__global__ void k_edge_agg(const int* __restrict__ src, const int* __restrict__ dst,
                           const int* __restrict__ emask, const float* __restrict__ ebuf,
                           const float* __restrict__ den, const float* __restrict__ hh,
                           float* __restrict__ out, int Ereal, int Nn, int co)
{
    int t = blockIdx.x * blockDim.x + threadIdx.x;
    int Et = Ereal + Nn;
    if (t >= Et * NHEADS) return;
    int e = t / NHEADS, h = t % NHEADS;
    int s, d, m;
    if (e < Ereal) { s = src[e]; d = dst[e]; m = emask[e]; }
    else           { s = d = e - Ereal; m = 1; }
    if (!m) return;
    float alpha = ebuf[t] / fmaxf(den[d * NHEADS + h], 1e-16f);
    if (alpha == 0.f) return;
    const float* hv = hh + ((size_t)s * NHEADS + h) * co;
    float*       ov = out + ((size_t)d * NHEADS + h) * co;
    for (int c = 0; c < co; ++c) atomicAdd(&ov[c], alpha * hv[c]);
}

// head mean + bias
__global__ void k_head_mean(const float* __restrict__ out, const float* __restrict__ gb,
                            float* __restrict__ h2, int Nn, int co)
{
    int t = blockIdx.x * blockDim.x + threadIdx.x;
    if (t >= Nn * co) return;
    int n = t / co, c = t % co;
    float s = 0.f;
    for (int h = 0; h < NHEADS; ++h) s += out[((size_t)n * NHEADS + h) * co + c];
    h2[t] = s * (1.0f / NHEADS) + gb[c];
}

// global mean pool per graph
__global__ void k_gmean(const float* __restrict__ h2, float* __restrict__ gm, int npg, int co)
{
    int t = blockIdx.x * blockDim.x + threadIdx.x;
    if (t >= B_GR * co) return;
    int b = t / co, c = t % co;
    float s = 0.f;
    for (int i = 0; i < npg; ++i) s += h2[((size_t)b * npg + i) * co + c];
    gm[t] = s / (float)npg;
}

__global__ void k_leaky(float* p, int n, float slope) {
    int t = blockIdx.x * blockDim.x + threadIdx.x;
    if (t < n) { float v = p[t]; p[t] = v >= 0.f ? v : slope * v; }
}

// GraphConv aggregate for pooling score: agg[dst,:] += h[src,:]
__global__ void k_pool_agg(const int* __restrict__ src, const int* __restrict__ dst,
                           const int* __restrict__ emask, const float* __restrict__ h,
                           float* __restrict__ agg, int Ereal, int co)
{
    int e = blockIdx.x * blockDim.x + threadIdx.x;
    if (e >= Ereal || !emask[e]) return;
    const float* hv = h + (size_t)src[e] * co;
    float*       av = agg + (size_t)dst[e] * co;
    for (int c = 0; c < co; ++c) atomicAdd(&av[c], hv[c]);
}

__global__ void k_pool_score(const float* __restrict__ agg, const float* __restrict__ h,
                             const float* __restrict__ relW, const float* __restrict__ relb,
                             const float* __restrict__ rootW, float* __restrict__ score,
                             int Nn, int co)
{
    int n = blockIdx.x * blockDim.x + threadIdx.x;
    if (n >= Nn) return;
    const float* av = agg + (size_t)n * co;
    const float* hv = h + (size_t)n * co;
    float s = relb[0];
    for (int c = 0; c < co; ++c) s += av[c] * relW[c] + hv[c] * rootW[c];
    score[n] = s;
}

// per-graph bitonic top-k (n = 256 or 128, blockDim == n)
__global__ void k_topk(const float* __restrict__ score, int* __restrict__ perm, int n, int k)
{
    __shared__ float ss[256];
    __shared__ int   si[256];
    int b = blockIdx.x, t = threadIdx.x;
    ss[t] = score[b * n + t]; si[t] = t;
    __syncthreads();
    for (int sz = 2; sz <= n; sz <<= 1) {
        for (int j = sz >> 1; j > 0; j >>= 1) {
            int ixj = t ^ j;
            if (ixj > t) {
                float a = ss[t], c = ss[ixj]; int ia = si[t], ic = si[ixj];
                bool aFirst = (a > c) || (a == c && ia < ic);  // desc, stable
                bool desc = ((t & sz) == 0);
                if (desc ? !aFirst : aFirst) { ss[t] = c; ss[ixj] = a; si[t] = ic; si[ixj] = ia; }
            }
            __syncthreads();
        }
    }
    if (t < k) perm[b * k + t] = b * n + si[t];
}

__global__ void k_scatter_pos(const int* __restrict__ perm, int* __restrict__ pos, int nk) {
    int t = blockIdx.x * blockDim.x + threadIdx.x; if (t < nk) pos[perm[t]] = t;
}

// h_new[j,:] = h_old[perm[j],:] * tanh(score[perm[j]])
__global__ void k_gather_gate(const float* __restrict__ h, const float* __restrict__ score,
                              const int* __restrict__ perm, float* __restrict__ out,
                              int nk, int co)
{
    int t = blockIdx.x * blockDim.x + threadIdx.x;
    if (t >= nk * co) return;
    int j = t / co, c = t % co;
    int pidx = perm[j];
    out[t] = h[(size_t)pidx * co + c] * tanhf(score[pidx]);
}

__global__ void k_remap(const int* __restrict__ pos, int* __restrict__ src,
                        int* __restrict__ dst, int* __restrict__ emask, int Ereal)
{
    int e = blockIdx.x * blockDim.x + threadIdx.x;
    if (e >= Ereal) return;
    int s2 = pos[src[e]], d2 = pos[dst[e]];
    int m = emask[e] && (s2 >= 0) && (d2 >= 0);
    emask[e] = m;
    src[e] = m ? s2 : 0;
    dst[e] = m ? d2 : 0;
}

// ==================================================================
extern "C" void kernel_launch(void* const* d_in, const int* in_sizes, int n_in,
                              void* d_out, int out_size, void* d_ws, size_t ws_size,
                              hipStream_t stream)
{
    (void)in_sizes; (void)n_in; (void)out_size; (void)ws_size;
    const float* x     = (const float*)d_in[0];
    const int*   eidx  = (const int*)  d_in[1];
    const float* esm   = (const float*)d_in[2];
    const float* cnn_W = (const float*)d_in[3];
    const float* cnn_b = (const float*)d_in[4];
    const float* bn_g  = (const float*)d_in[5];
    const float* bn_be = (const float*)d_in[6];
    const float* bn_m  = (const float*)d_in[7];
    const float* bn_v  = (const float*)d_in[8];
    const float* fc_W  = (const float*)d_in[9];
    const float* fc_b  = (const float*)d_in[10];
    const float* esm_W = (const float*)d_in[11];
    const float* esm_b = (const float*)d_in[12];
    const float *gat_W[3], *att_s[3], *att_d[3], *gat_b[3], *mlp_W[3], *mlp_b[3];
    const float *rel_W[2], *rel_b[2], *root_W[2];
    int p = 13;
    for (int i = 0; i < 3; ++i) {
        gat_W[i] = (const float*)d_in[p++];
        att_s[i] = (const float*)d_in[p++];
        att_d[i] = (const float*)d_in[p++];
        gat_b[i] = (const float*)d_in[p++];
        mlp_W[i] = (const float*)d_in[p++];
        mlp_b[i] = (const float*)d_in[p++];
        if (i < 2) {
            rel_W[i]  = (const float*)d_in[p++];
            rel_b[i]  = (const float*)d_in[p++];
            root_W[i] = (const float*)d_in[p++];
        }
    }

    // ---- carve workspace
    char* wp = (char*)d_ws;
    auto carve = [&](size_t bytes) -> void* {
        void* r = (void*)wp; wp += (bytes + 255) & ~(size_t)255; return r;
    };
    float*    hh     = (float*)   carve((size_t)8388608 * 4);   // Nn*8*co (const 8.4M)
    float*    aggout = (float*)   carve((size_t)8388608 * 4);
    float*    hA     = (float*)   carve((size_t)524288 * 4);    // layer input
    float*    h2     = (float*)   carve((size_t)1048576 * 4);   // conv output (head-mean)
    float*    asrc   = (float*)   carve((size_t)65536 * 4);
    float*    adst   = (float*)   carve((size_t)65536 * 4);
    float*    ebuf   = (float*)   carve((size_t)(NE + 8192) * NHEADS * 4);
    unsigned* mxk    = (unsigned*)carve((size_t)65536 * 4);
    float*    den    = (float*)   carve((size_t)65536 * 4);
    float*    agg    = (float*)   carve((size_t)1048576 * 4);
    float*    score  = (float*)   carve((size_t)8192 * 4);
    float*    fea    = (float*)   carve((size_t)B_GR * 512 * 4);
    float*    gmean  = (float*)   carve((size_t)B_GR * 512 * 4);
    int*      perm   = (int*)     carve((size_t)4096 * 4);
    int*      pos    = (int*)     carve((size_t)8192 * 4);
    int*      srcb   = (int*)     carve((size_t)NE * 4);
    int*      dstb   = (int*)     carve((size_t)NE * 4);
    int*      emaskb = (int*)     carve((size_t)NE * 4);

    auto cdiv = [](int a, int b) { return (a + b - 1) / b; };
    auto gemm = [&](const float* A, const float* W, float* C, int M, int N, int K,
                    const float* bias, int epi, float slope,
                    const float* g = nullptr, const float* be = nullptr,
                    const float* mn = nullptr, const float* vr = nullptr) {
        // each wave: 16x64; each block (4 waves): 16x256
        dim3 grid(cdiv(N, 256), M / 16);
        wmma_gemm_f32<<<grid, 128, 0, stream>>>(A, W, C, M, N, K, bias, g, be, mn, vr, epi, slope);
    };

    // ---- h0 = leaky(BN(x @ cnn_W + cnn_b), 0.01)   [8192 x 64]
    gemm(x, cnn_W, hA, 8192, 64, 128, cnn_b, EPI_BN_LEAKY, 0.01f, bn_g, bn_be, bn_m, bn_v);
    k_fill_f32<<<cdiv(B_GR * 512, 256), 256, 0, stream>>>(fea, 0.f, B_GR * 512);
    k_copy_edges<<<cdiv(NE, 256), 256, 0, stream>>>(eidx, srcb, dstb, emaskb);

    int n = NPG0, Nn = B_GR * NPG0;
    const int ci_[3] = {64, 128, 256}, co_[3] = {128, 256, 512};
    for (int i = 0; i < 3; ++i) {
        const int ci = ci_[i], co = co_[i];
        // hh = h @ gat_W : [Nn, 8*co]
        gemm(hA, gat_W[i], hh, Nn, 8 * co, ci, nullptr, EPI_STORE, 0.f);
        k_att_coeff<<<cdiv(Nn * NHEADS, 256), 256, 0, stream>>>(hh, att_s[i], att_d[i], asrc, adst, Nn, co);
        const int Et = NE + Nn;
        k_fill_key<<<cdiv(Nn * NHEADS, 256), 256, 0, stream>>>(mxk, NEG_BIG, Nn * NHEADS);
        k_fill_f32<<<cdiv(Nn * NHEADS, 256), 256, 0, stream>>>(den, 0.f, Nn * NHEADS);
        k_fill_f32<<<cdiv(Nn * NHEADS * co, 256), 256, 0, stream>>>(aggout, 0.f, Nn * NHEADS * co);
        k_edge_scores<<<cdiv(Et * NHEADS, 256), 256, 0, stream>>>(srcb, dstb, emaskb, asrc, adst, ebuf, mxk, NE, Nn);
        k_edge_w<<<cdiv(Et * NHEADS, 256), 256, 0, stream>>>(srcb, dstb, emaskb, mxk, ebuf, den, NE, Nn);
        k_edge_agg<<<cdiv(Et * NHEADS, 256), 256, 0, stream>>>(srcb, dstb, emaskb, ebuf, den, hh, aggout, NE, Nn, co);
        k_head_mean<<<cdiv(Nn * co, 256), 256, 0, stream>>>(aggout, gat_b[i], h2, Nn, co);
        // fea += mean_pool(h2) @ mlp_W + mlp_b
        k_gmean<<<cdiv(B_GR * co, 256), 256, 0, stream>>>(h2, gmean, n, co);
        gemm(gmean, mlp_W[i], fea, B_GR, 512, co, mlp_b[i], EPI_ACCUM, 0.f);
        if (i < 2) {
            // SAGPool(0.5)
            k_leaky<<<cdiv(Nn * co, 256), 256, 0, stream>>>(h2, Nn * co, 0.01f);   // h = leaky(h2)
            k_fill_f32<<<cdiv(Nn * co, 256), 256, 0, stream>>>(agg, 0.f, Nn * co);
            k_pool_agg<<<cdiv(NE, 256), 256, 0, stream>>>(srcb, dstb, emaskb, h2, agg, NE, co);
            k_pool_score<<<cdiv(Nn, 256), 256, 0, stream>>>(agg, h2, rel_W[i], rel_b[i], root_W[i], score, Nn, co);
            const int k = n / 2;
            k_topk<<<B_GR, n, 0, stream>>>(score, perm, n, k);
            k_fill_i32<<<cdiv(Nn, 256), 256, 0, stream>>>(pos, -1, Nn);
            k_scatter_pos<<<cdiv(B_GR * k, 256), 256, 0, stream>>>(perm, pos, B_GR * k);
            k_gather_gate<<<cdiv(B_GR * k * co, 256), 256, 0, stream>>>(h2, score, perm, hA, B_GR * k, co);
            k_remap<<<cdiv(NE, 256), 256, 0, stream>>>(pos, srcb, dstb, emaskb, NE);
            n = k; Nn = B_GR * k;
        }
    }

    // out = fea @ fc_W + fc_b + esm @ esm_W + esm_b
    float* out = (float*)d_out;
    gemm(fea, fc_W, out, B_GR, 512, 512, fc_b, EPI_STORE, 0.f);
    gemm(esm, esm_W, out, B_GR, 512, 1280, esm_b, EPI_ACCUM, 0.f);
}